// GCNBlock_68925635166995
// MI455X (gfx1250) — compile-verified
//
#include <hip/hip_runtime.h>

typedef __attribute__((ext_vector_type(2))) float v2f;
typedef __attribute__((ext_vector_type(8))) float v8f;

#define D_FEAT 64
#define LDS_PAD 68                 // 68*4 B row stride: 16B aligned, bank shift 4/row -> conflict-free
#define ROWS_PER_WAVE 16
#define WAVES_PER_BLOCK 8
#define ROWS_PER_BLOCK (ROWS_PER_WAVE * WAVES_PER_BLOCK)

// 32-bit LDS byte offset of a generic pointer into __shared__ memory
// (generic -> addrspace(3) cast; low 32 bits of an LDS-aperture address are
// the LDS offset per the CDNA5 aperture rules).
__device__ __forceinline__ unsigned lds_offset(const void* p) {
    return (unsigned)(uintptr_t)(const __attribute__((address_space(3))) void*)p;
}

// Async direct global->LDS 128-bit copy (CDNA5 GLOBAL_LOAD_ASYNC_TO_LDS_B128,
// tracked by ASYNCcnt). ldsDst: per-lane LDS byte offset; gaddr: per-lane
// 64-bit global address.
__device__ __forceinline__ void async_g2lds_b128(unsigned ldsDst, unsigned long long gaddr) {
    asm volatile("global_load_async_to_lds_b128 %0, %1, off"
                 :: "v"(ldsDst), "v"(gaddr)
                 : "memory");
}

__device__ __forceinline__ void wait_asynccnt0() {
    asm volatile("s_wait_asynccnt 0" ::: "memory");
}

// ---------------------------------------------------------------------------
// Zero the output accumulator (harness poisons d_out with 0xAA).
// ---------------------------------------------------------------------------
__global__ __launch_bounds__(256) void gcn_zero_f4(float4* __restrict__ p, int n4) {
    int i = blockIdx.x * blockDim.x + threadIdx.x;
    if (i < n4) p[i] = make_float4(0.f, 0.f, 0.f, 0.f);
}

// ---------------------------------------------------------------------------
// h = x @ W^T via V_WMMA_F32_16X16X4_F32.
//   A tile: 16 nodes x 4 dims (from LDS-staged x tile)
//   B tile: 4 dims x 16 outs  (B[k][n] = W[n][k], from LDS-staged W)
//   D/C   : 16x16 f32, 4 n-tiles cover D_OUT=64
// Staging uses async direct-to-LDS b128 copies (ASYNCcnt), no VGPR round trip.
// ---------------------------------------------------------------------------
__global__ __launch_bounds__(256) void gcn_gemm_wmma(
    const float* __restrict__ x, const float* __restrict__ W,
    float* __restrict__ h, int nNodes)
{
    __shared__ float sW[D_FEAT * LDS_PAD];
    __shared__ float sX[WAVES_PER_BLOCK][ROWS_PER_WAVE * LDS_PAD];

    const int tid  = threadIdx.x;
    const int lane = tid & 31;
    const int wave = tid >> 5;
    const int rowBase = blockIdx.x * ROWS_PER_BLOCK + wave * ROWS_PER_WAVE;

    // Stage W (64x64 f32 = 1024 float4) into padded LDS: 256 threads x 4 async b128.
    #pragma unroll
    for (int i = 0; i < 4; ++i) {
        int f  = i * 256 + tid;        // float4 index
        int r  = f >> 4;               // 16 float4 per 64-float row
        int c4 = f & 15;
        unsigned dst = lds_offset(&sW[r * LDS_PAD + c4 * 4]);
        async_g2lds_b128(dst, (unsigned long long)(uintptr_t)(W + (size_t)f * 4));
    }

    // Stage this wave's 16x64 x tile (256 float4 / 32 lanes = 8 async b128 each).
    // nNodes is a multiple of 16, so a tile is either fully valid or fully skipped
    // (wave-uniform branch keeps things simple; EXEC all-ones inside).
    if (rowBase < nNodes) {
        const float* xbase = x + (size_t)rowBase * D_FEAT;
        #pragma unroll
        for (int i = 0; i < 8; ++i) {
            int f  = i * 32 + lane;
            int r  = f >> 4;
            int c4 = f & 15;
            unsigned dst = lds_offset(&sX[wave][r * LDS_PAD + c4 * 4]);
            async_g2lds_b128(dst, (unsigned long long)(uintptr_t)(xbase + (size_t)f * 4));
        }
    }
    wait_asynccnt0();      // this wave's async copies are in LDS
    __syncthreads();       // ... and visible to all waves in the block

    if (rowBase >= nNodes) return;   // uniform whole-wave skip: EXEC stays all-ones

    const int half = lane >> 4;      // 0 or 1
    const int lidx = lane & 15;      // M (for A/D) or N (for B/D)

    v8f acc0 = {}, acc1 = {}, acc2 = {}, acc3 = {};
    const float* xt = &sX[wave][0];

    #pragma unroll
    for (int k0 = 0; k0 < D_FEAT; k0 += 4) {
        const int kk = k0 + half * 2;
        v2f a  = *(const v2f*)&xt[lidx * LDS_PAD + kk];
        v2f b0 = *(const v2f*)&sW[( 0 + lidx) * LDS_PAD + kk];
        v2f b1 = *(const v2f*)&sW[(16 + lidx) * LDS_PAD + kk];
        v2f b2 = *(const v2f*)&sW[(32 + lidx) * LDS_PAD + kk];
        v2f b3 = *(const v2f*)&sW[(48 + lidx) * LDS_PAD + kk];
        acc0 = __builtin_amdgcn_wmma_f32_16x16x4_f32(false, a, false, b0, (short)0, acc0, false, false);
        acc1 = __builtin_amdgcn_wmma_f32_16x16x4_f32(false, a, false, b1, (short)0, acc1, false, false);
        acc2 = __builtin_amdgcn_wmma_f32_16x16x4_f32(false, a, false, b2, (short)0, acc2, false, false);
        acc3 = __builtin_amdgcn_wmma_f32_16x16x4_f32(false, a, false, b3, (short)0, acc3, false, false);
    }

    // Write back D tiles: lane holds N=lidx; VGPR v holds row M = v + 8*half.
    float* out = h + (size_t)rowBase * D_FEAT;
    #pragma unroll
    for (int v = 0; v < 8; ++v) {
        const int m = v + half * 8;
        out[(size_t)m * D_FEAT +  0 + lidx] = acc0[v];
        out[(size_t)m * D_FEAT + 16 + lidx] = acc1[v];
        out[(size_t)m * D_FEAT + 32 + lidx] = acc2[v];
        out[(size_t)m * D_FEAT + 48 + lidx] = acc3[v];
    }
}

// ---------------------------------------------------------------------------
// Edge gather/scale/scatter-add. One wave per edge; 2 floats per lane.
// 32 edges' metadata loaded coalesced per wave, broadcast with shuffles.
// unsafeAtomicAdd -> native global_atomic_add_f32 (no-return, STOREcnt).
// h and delta (25.6 MB each) are L2-resident on MI455X (192 MB L2).
// ---------------------------------------------------------------------------
__global__ __launch_bounds__(256) void gcn_edge_scatter(
    const float* __restrict__ h, const float* __restrict__ ew,
    const int* __restrict__ src, const int* __restrict__ tgt,
    float* __restrict__ delta, int nEdges)
{
    const int lane   = threadIdx.x & 31;
    const int wave   = (blockIdx.x * blockDim.x + threadIdx.x) >> 5;
    const int nWaves = (gridDim.x * blockDim.x) >> 5;

    for (int base = wave * 32; base < nEdges; base += nWaves * 32) {
        int s = 0, t = 0;
        float w = 0.f;
        const int e = base + lane;
        if (e < nEdges) { s = src[e]; t = tgt[e]; w = ew[e]; }
        const int cnt = (nEdges - base < 32) ? (nEdges - base) : 32;

        for (int j = 0; j < cnt; ++j) {
            const int   sj = __shfl(s, j, 32);
            const int   tj = __shfl(t, j, 32);
            const float wj = __shfl(w, j, 32);

            // Prefetch next gathered row into cache (global_prefetch_b8).
            if (j + 1 < cnt) {
                const int sn = __shfl(s, j + 1, 32);
                __builtin_prefetch(h + (size_t)sn * D_FEAT + lane * 2, 0, 3);
            }

            const float2 hv = ((const float2*)(h + (size_t)sj * D_FEAT))[lane];
            float* dp = delta + (size_t)tj * D_FEAT + lane * 2;
            unsafeAtomicAdd(dp,     hv.x * wj);
            unsafeAtomicAdd(dp + 1, hv.y * wj);
        }
    }
}

// ---------------------------------------------------------------------------
// Launch: zero delta -> WMMA GEMM into workspace -> edge scatter into d_out.
// Inputs: [0]=x f32(100000*64), [1]=W f32(64*64), [2]=edge_weights f32(1e6),
//         [3]=source i32(1e6), [4]=target i32(1e6), [5]=num_nodes i32(1)
// ---------------------------------------------------------------------------
extern "C" void kernel_launch(void* const* d_in, const int* in_sizes, int n_in,
                              void* d_out, int out_size, void* d_ws, size_t ws_size,
                              hipStream_t stream) {
    const float* x   = (const float*)d_in[0];
    const float* W   = (const float*)d_in[1];
    const float* ew  = (const float*)d_in[2];
    const int*   src = (const int*)d_in[3];
    const int*   tgt = (const int*)d_in[4];

    const int nNodes = in_sizes[0] / D_FEAT;   // 100000
    const int nEdges = in_sizes[2];            // 1000000

    float* delta = (float*)d_out;
    float* h     = (float*)d_ws;               // 100000*64*4 = 25.6 MB scratch

    const int n4 = (nNodes * D_FEAT) / 4;
    gcn_zero_f4<<<(n4 + 255) / 256, 256, 0, stream>>>((float4*)delta, n4);

    const int gemmBlocks = (nNodes + ROWS_PER_BLOCK - 1) / ROWS_PER_BLOCK;
    gcn_gemm_wmma<<<gemmBlocks, 256, 0, stream>>>(x, W, h, nNodes);

    gcn_edge_scatter<<<2048, 256, 0, stream>>>(h, ew, src, tgt, delta, nEdges);
}